// PolynomialLearningRule_69269232550108
// MI455X (gfx1250) — compile-verified
//
#include <hip/hip_runtime.h>

typedef float v2f __attribute__((ext_vector_type(2)));
typedef float v8f __attribute__((ext_vector_type(8)));
typedef int   v4i __attribute__((vector_size(16)));   // matches builtin param type

#define BDIM 8192
#define DDIM 512
#define KC   32        // K-chunk staged through LDS
#define NKCHUNK (BDIM / KC)
#define NPART 12       // column-sum vectors: y,y2,y3,s,sy,sy2,s2,s2y,s3,x,x2,x3
#define NCHUNK 64      // b-chunks for colsum partials

#define AS_GLOBAL __attribute__((address_space(1)))
#define AS_LDS    __attribute__((address_space(3)))

#if defined(__HIP_DEVICE_COMPILE__) && __has_builtin(__builtin_amdgcn_global_load_async_to_lds_b128)
#define USE_ASYNC_LDS 1
#else
#define USE_ASYNC_LDS 0
#endif

// ---------------------------------------------------------------------------
// Kernel 1: partial column sums over b for the 12 power/product vectors.
// grid (2, 64) x 256 threads; thread -> one output column o, one b-chunk.
// ---------------------------------------------------------------------------
__global__ __launch_bounds__(256) void colsum_kernel(const float* __restrict__ x,
                                                     const float* __restrict__ s,
                                                     const float* __restrict__ y,
                                                     float* __restrict__ part) {
  const int o = blockIdx.x * 256 + threadIdx.x;        // 0..511
  const int c = blockIdx.y;                            // 0..63
  const int b0 = c * (BDIM / NCHUNK);                  // 128 rows per chunk
  float a0 = 0.f, a1 = 0.f, a2 = 0.f, a3 = 0.f, a4 = 0.f, a5 = 0.f;
  float a6 = 0.f, a7 = 0.f, a8 = 0.f, a9 = 0.f, a10 = 0.f, a11 = 0.f;
  for (int b = b0; b < b0 + (BDIM / NCHUNK); ++b) {
    const float yv = y[b * DDIM + o];
    const float sv = s[b * DDIM + o];
    const float xv = x[b * DDIM + o];
    const float y2 = yv * yv, s2 = sv * sv, x2 = xv * xv;
    a0 += yv;       a1 += y2;       a2 += y2 * yv;
    a3 += sv;       a4 += sv * yv;  a5 += sv * y2;
    a6 += s2;       a7 += s2 * yv;  a8 += s2 * sv;
    a9 += xv;       a10 += x2;      a11 += x2 * xv;
  }
  float* p = part + (size_t)c * NPART * DDIM + o;
  p[0 * DDIM] = a0;  p[1 * DDIM] = a1;  p[2 * DDIM] = a2;  p[3 * DDIM] = a3;
  p[4 * DDIM] = a4;  p[5 * DDIM] = a5;  p[6 * DDIM] = a6;  p[7 * DDIM] = a7;
  p[8 * DDIM] = a8;  p[9 * DDIM] = a9;  p[10 * DDIM] = a10; p[11 * DDIM] = a11;
}

// ---------------------------------------------------------------------------
// Kernel 2: reduce the 64 partials -> finals[12][512].
// grid 24 x 256 threads (6144 outputs).
// ---------------------------------------------------------------------------
__global__ __launch_bounds__(256) void reduce_kernel(const float* __restrict__ part,
                                                     float* __restrict__ fin) {
  const int idx = blockIdx.x * 256 + threadIdx.x;      // 0..6143
  float acc = 0.f;
  for (int c = 0; c < NCHUNK; ++c) acc += part[(size_t)c * NPART * DDIM + idx];
  fin[idx] = acc;
}

// ---------------------------------------------------------------------------
// Kernel 3: the 7 remaining true GEMMs via V_WMMA_F32_16X16X4_F32, with
// double-buffered GLOBAL_LOAD_ASYNC_TO_LDS_B128 staging (ASYNCcnt-tracked).
// grid (16,16) blocks of 128 threads = 4 waves in a 2x2 tile arrangement,
// each wave owns one 16x16 output tile with 7 f32 accumulators.
// ---------------------------------------------------------------------------
__global__ __launch_bounds__(128) void gemm_kernel(const float* __restrict__ x,
                                                   const float* __restrict__ s,
                                                   const float* __restrict__ y,
                                                   const float* __restrict__ w,
                                                   const float* __restrict__ pw,
                                                   const float* __restrict__ fin,
                                                   float* __restrict__ out) {
  __shared__ float Ys[2][KC][40];  // stride 40 floats: lanes 0-15 vs 16-31 hit
  __shared__ float Ss[2][KC][40];  // disjoint bank groups (2 rows = 80 = 16 mod 64)
  __shared__ float Xs[2][KC][40];
  __shared__ float sPar[35];
  __shared__ float sSy[9][32];
  __shared__ float sSx[3][32];

  const int tid = threadIdx.x;
  const int oBase = blockIdx.y * 32;
  const int iBase = blockIdx.x * 32;

  // stage epilogue data (params + broadcast column-sum slices)
  if (tid < 35) sPar[tid] = pw[tid];
  for (int j = tid; j < 288; j += 128) {
    const int r = j >> 5, c = j & 31;
    sSy[r][c] = fin[r * DDIM + oBase + c];
  }
  for (int j = tid; j < 96; j += 128) {
    const int r = j >> 5, c = j & 31;
    sSx[r][c] = fin[(9 + r) * DDIM + iBase + c];
  }

  const int wv = tid >> 5, lane = tid & 31;
  const int waveRow = wv >> 1, waveCol = wv & 1;
  const int h = lane >> 4, m = lane & 15;
  const int oc = waveRow * 16 + m;      // column within the 32-wide o tile
  const int ic = waveCol * 16 + m;      // column within the 32-wide i tile

  // cooperative-load indices: 4 threads per row, 8 floats (2 x b128) each
  const int lr = tid >> 2;
  const int lc = (tid & 3) * 8;

  v8f g0 = {}, g1 = {}, g2 = {}, g3 = {}, g4 = {}, g5 = {}, g6 = {};

#if USE_ASYNC_LDS
  // Issue the 6 async global->LDS b128 copies for one K-chunk into buffer `buf`.
  auto issue_chunk = [&](int buf, int kb) {
    const float* yp = y + (size_t)(kb + lr) * DDIM + oBase + lc;
    const float* sp = s + (size_t)(kb + lr) * DDIM + oBase + lc;
    const float* xp = x + (size_t)(kb + lr) * DDIM + iBase + lc;
    __builtin_amdgcn_global_load_async_to_lds_b128(
        (AS_GLOBAL v4i*)yp, (AS_LDS v4i*)&Ys[buf][lr][lc], 0, 0);
    __builtin_amdgcn_global_load_async_to_lds_b128(
        (AS_GLOBAL v4i*)(yp + 4), (AS_LDS v4i*)&Ys[buf][lr][lc + 4], 0, 0);
    __builtin_amdgcn_global_load_async_to_lds_b128(
        (AS_GLOBAL v4i*)sp, (AS_LDS v4i*)&Ss[buf][lr][lc], 0, 0);
    __builtin_amdgcn_global_load_async_to_lds_b128(
        (AS_GLOBAL v4i*)(sp + 4), (AS_LDS v4i*)&Ss[buf][lr][lc + 4], 0, 0);
    __builtin_amdgcn_global_load_async_to_lds_b128(
        (AS_GLOBAL v4i*)xp, (AS_LDS v4i*)&Xs[buf][lr][lc], 0, 0);
    __builtin_amdgcn_global_load_async_to_lds_b128(
        (AS_GLOBAL v4i*)(xp + 4), (AS_LDS v4i*)&Xs[buf][lr][lc + 4], 0, 0);
  };

  issue_chunk(0, 0);
  for (int c = 0; c < NKCHUNK; ++c) {
    const int p = c & 1;
    if (c + 1 < NKCHUNK) {
      issue_chunk(p ^ 1, (c + 1) * KC);
      // wait for own chunk-c copies (first 6, in-order); 6 prefetch copies remain
      asm volatile("s_wait_asynccnt 0x6" ::: "memory");
    } else {
      asm volatile("s_wait_asynccnt 0x0" ::: "memory");
    }
    __syncthreads();   // cross-wave: everyone's chunk-c data is in LDS
#else
  for (int c = 0; c < NKCHUNK; ++c) {
    const int p = 0;
    const int kb = c * KC;
    const float* yp = y + (size_t)(kb + lr) * DDIM + oBase + lc;
    const float* sp = s + (size_t)(kb + lr) * DDIM + oBase + lc;
    const float* xp = x + (size_t)(kb + lr) * DDIM + iBase + lc;
    float4 yv0 = *(const float4*)yp, yv1 = *(const float4*)(yp + 4);
    float4 sv0 = *(const float4*)sp, sv1 = *(const float4*)(sp + 4);
    float4 xv0 = *(const float4*)xp, xv1 = *(const float4*)(xp + 4);
    *(float4*)&Ys[0][lr][lc] = yv0; *(float4*)&Ys[0][lr][lc + 4] = yv1;
    *(float4*)&Ss[0][lr][lc] = sv0; *(float4*)&Ss[0][lr][lc + 4] = sv1;
    *(float4*)&Xs[0][lr][lc] = xv0; *(float4*)&Xs[0][lr][lc + 4] = xv1;
    __syncthreads();
#endif

#pragma unroll
    for (int kk = 0; kk < KC; kk += 4) {
      const int ka = kk + 2 * h;   // A/B: lanes 0-15 hold K=kk,kk+1; 16-31 K=kk+2,kk+3
      const float ya = Ys[p][ka][oc], yb = Ys[p][ka + 1][oc];
      const float sa = Ss[p][ka][oc], sb = Ss[p][ka + 1][oc];
      const float xa = Xs[p][ka][ic], xb = Xs[p][ka + 1][ic];
      v2f A1 = {ya, yb};                 // y
      v2f A2 = {ya * ya, yb * yb};       // y^2
      v2f A4 = {sa, sb};                 // s
      v2f A5 = {sa * ya, sb * yb};       // s*y
      v2f A7 = {sa * sa, sb * sb};       // s^2
      v2f B1 = {xa, xb};                 // x
      v2f B2 = {xa * xa, xb * xb};       // x^2
      g0 = __builtin_amdgcn_wmma_f32_16x16x4_f32(false, A1, false, B1, (short)0, g0, false, false);
      g1 = __builtin_amdgcn_wmma_f32_16x16x4_f32(false, A2, false, B1, (short)0, g1, false, false);
      g2 = __builtin_amdgcn_wmma_f32_16x16x4_f32(false, A4, false, B1, (short)0, g2, false, false);
      g3 = __builtin_amdgcn_wmma_f32_16x16x4_f32(false, A5, false, B1, (short)0, g3, false, false);
      g4 = __builtin_amdgcn_wmma_f32_16x16x4_f32(false, A7, false, B1, (short)0, g4, false, false);
      g5 = __builtin_amdgcn_wmma_f32_16x16x4_f32(false, A1, false, B2, (short)0, g5, false, false);
      g6 = __builtin_amdgcn_wmma_f32_16x16x4_f32(false, A4, false, B2, (short)0, g6, false, false);
    }
    __syncthreads();   // buffer p free for the next prefetch
  }

  // Epilogue: C/D layout: VGPR r, half h -> M = r + 8h, N = lane&15.
  const int i0 = iBase + ic;
  const float invB = 1.0f / (float)BDIM;
#pragma unroll
  for (int r = 0; r < 8; ++r) {
    const int ml = waveRow * 16 + r + 8 * h;   // row within block o tile
    const int o = oBase + ml;
    const float wvv = w[(size_t)o * DDIM + i0];
    const float w2 = wvv * wvv, w3 = w2 * wvv;

    // t0: constant correlation = B
    float total = (sPar[0] + sPar[1] * wvv + sPar[2] * w2 + sPar[3] * w3) * (float)BDIM;
    // ex=0 terms (broadcast over i): pairs y..s^3
    total += (sPar[4] + sPar[5] * wvv + sPar[6] * w2) * sSy[0][ml];   // y
    total += (sPar[7] + sPar[8] * wvv) * sSy[1][ml];                  // y^2
    total += sPar[9] * sSy[2][ml];                                    // y^3
    total += (sPar[10] + sPar[11] * wvv + sPar[12] * w2) * sSy[3][ml];// s
    total += (sPar[13] + sPar[14] * wvv) * sSy[4][ml];                // s*y
    total += sPar[15] * sSy[5][ml];                                   // s*y^2
    total += (sPar[16] + sPar[17] * wvv) * sSy[6][ml];                // s^2
    total += sPar[18] * sSy[7][ml];                                   // s^2*y
    total += sPar[19] * sSy[8][ml];                                   // s^3
    // es=ey=0, ex>=1 terms (broadcast over o)
    total += (sPar[20] + sPar[21] * wvv + sPar[22] * w2) * sSx[0][ic];// x
    total += (sPar[30] + sPar[31] * wvv) * sSx[1][ic];                // x^2
    total += sPar[34] * sSx[2][ic];                                   // x^3
    // true GEMM terms
    total += (sPar[23] + sPar[24] * wvv) * g0[r];   // y  . x
    total += sPar[25] * g1[r];                      // y^2. x
    total += (sPar[26] + sPar[27] * wvv) * g2[r];   // s  . x
    total += sPar[28] * g3[r];                      // sy . x
    total += sPar[29] * g4[r];                      // s^2. x
    total += sPar[32] * g5[r];                      // y  . x^2
    total += sPar[33] * g6[r];                      // s  . x^2

    out[(size_t)o * DDIM + i0] = invB * total;
  }
}

// ---------------------------------------------------------------------------
// Kernel 4: delta_b from y-power column sums. grid 2 x 256.
// ---------------------------------------------------------------------------
__global__ __launch_bounds__(256) void bias_kernel(const float* __restrict__ fin,
                                                   const float* __restrict__ b,
                                                   const float* __restrict__ pb,
                                                   float* __restrict__ outb) {
  const int o = blockIdx.x * 256 + threadIdx.x;
  const float bv = b[o], b2 = bv * bv, b3 = b2 * bv;
  const float S1 = fin[0 * DDIM + o];   // sum y
  const float S2 = fin[1 * DDIM + o];   // sum y^2
  const float S3 = fin[2 * DDIM + o];   // sum y^3
  float total = (pb[0] + pb[1] * bv + pb[2] * b2 + pb[3] * b3) * (float)BDIM;
  total += (pb[4] + pb[5] * bv + pb[6] * b2) * S1;
  total += (pb[7] + pb[8] * bv) * S2;
  total += pb[9] * S3;
  outb[o] = total * (1.0f / (float)BDIM);
}

extern "C" void kernel_launch(void* const* d_in, const int* in_sizes, int n_in,
                              void* d_out, int out_size, void* d_ws, size_t ws_size,
                              hipStream_t stream) {
  const float* x  = (const float*)d_in[0];
  const float* s  = (const float*)d_in[1];
  const float* y  = (const float*)d_in[2];
  const float* w  = (const float*)d_in[3];
  const float* b  = (const float*)d_in[4];
  const float* pw = (const float*)d_in[5];
  const float* pb = (const float*)d_in[6];
  float* out = (float*)d_out;

  float* part = (float*)d_ws;                         // [64][12][512]
  float* fin  = part + (size_t)NCHUNK * NPART * DDIM; // [12][512]

  colsum_kernel<<<dim3(2, NCHUNK), 256, 0, stream>>>(x, s, y, part);
  reduce_kernel<<<(NPART * DDIM) / 256, 256, 0, stream>>>(part, fin);
  gemm_kernel<<<dim3(DDIM / 32, DDIM / 32), 128, 0, stream>>>(x, s, y, w, pw, fin, out);
  bias_kernel<<<2, 256, 0, stream>>>(fin, b, pb, out + DDIM * DDIM);
}